// PointRender2D_68332929679531
// MI455X (gfx1250) — compile-verified
//
#include <hip/hip_runtime.h>
#include <stdint.h>

// Problem constants (fixed by the reference).
#define HEIGHT 2048
#define WIDTH  2048
#define RADIUS 4
#define WORDS_PER_ROW (WIDTH / 32)                 // 64 u32 words per row
#define MASK_WORDS    (HEIGHT * WORDS_PER_ROW)     // 131072 words
#define MASK_BYTES    (MASK_WORDS * 4)             // 512 KiB

// ---------------------------------------------------------------------------
// Kernel 1: zero the coverage bitmask (runs every call; scratch is poisoned).
// ---------------------------------------------------------------------------
__global__ void PR_clear_mask(uint4* __restrict__ m) {
    unsigned i = blockIdx.x * blockDim.x + threadIdx.x;
    if (i < MASK_WORDS / 4) m[i] = make_uint4(0u, 0u, 0u, 0u);
}

// ---------------------------------------------------------------------------
// Kernel 2: one thread per point; set covered bits (same math as reference:
// floor(), (px-cx)^2+(py-cy)^2 <= r^2, in-bounds). Per row the covered span
// touches at most 2 mask words -> <=2 atomicOr per row.
// ---------------------------------------------------------------------------
__global__ void PR_splat(const float* __restrict__ centers,
                         const float* __restrict__ vp,
                         int n, unsigned* __restrict__ mask) {
    int i = blockIdx.x * blockDim.x + threadIdx.x;
    if (i >= n) return;
    const float vx = vp[0], vy = vp[1], vw = vp[2], vh = vp[3];
    const float cx = (centers[2 * i]     - vx) / vw * (float)WIDTH;
    const float cy = (float)HEIGHT - (centers[2 * i + 1] - vy) / vh * (float)HEIGHT;
    const int   bx = (int)floorf(cx);
    const int   by = (int)floorf(cy);
    const float r2 = (float)(RADIUS * RADIUS);

    for (int dy = -RADIUS; dy <= RADIUS; ++dy) {
        int py = by + dy;
        if (py < 0 || py >= HEIGHT) continue;
        float fy  = (float)py - cy;
        float fy2 = fy * fy;
        int wlo = -1, whi = -1;
        unsigned blo = 0u, bhi = 0u;
        for (int dx = -RADIUS; dx <= RADIUS; ++dx) {
            int px = bx + dx;
            if (px < 0 || px >= WIDTH) continue;
            float fx = (float)px - cx;
            if (fx * fx + fy2 <= r2) {
                int      w = px >> 5;
                unsigned b = 1u << (px & 31);
                if (wlo < 0 || w == wlo) { wlo = w; blo |= b; }
                else                     { whi = w; bhi |= b; }
            }
        }
        unsigned* rowp = mask + (size_t)py * WORDS_PER_ROW;
        if (wlo >= 0) atomicOr(rowp + wlo, blo);
        if (whi >= 0) atomicOr(rowp + whi, bhi);
    }
}

// ---------------------------------------------------------------------------
// Kernel 3: one block per image row. Stage the row's 64 mask words (256 B)
// into LDS with the CDNA5 async global->LDS path (ASYNCcnt), then emit the
// 32 KiB row as fully coalesced float4 (global_store_b128) stores.
// ---------------------------------------------------------------------------
__global__ void PR_colorize(const unsigned* __restrict__ mask,
                            float4* __restrict__ out) {
    __shared__ __align__(16) unsigned smask[WORDS_PER_ROW];
    const unsigned tid = threadIdx.x;
    const unsigned row = blockIdx.x;

    if (tid < 16u) {
        // Per-lane 16-byte async copy: lanes 0..15 move 256 B of mask bits.
        unsigned lds_off = (unsigned)(unsigned long long)&smask[tid * 4u];
        unsigned goff    = row * (unsigned)(WORDS_PER_ROW * 4) + tid * 16u;
        asm volatile(
            "global_load_async_to_lds_b128 %0, %1, %2\n\t"
            "s_wait_asynccnt 0x0"
            :
            : "v"(lds_off), "v"(goff), "s"(mask)
            : "memory");
    }
    __syncthreads();

    const float4 red   = make_float4(1.f, 0.f, 0.f, 1.f);
    const float4 white = make_float4(1.f, 1.f, 1.f, 1.f);
    float4* o = out + (size_t)row * WIDTH;
#pragma unroll
    for (int k = 0; k < 8; ++k) {
        unsigned x     = (unsigned)k * 256u + tid;     // pixel column
        unsigned wbits = smask[x >> 5];                // wave-uniform LDS word
        o[x] = ((wbits >> (tid & 31u)) & 1u) ? red : white;
    }
}

// ---------------------------------------------------------------------------
// Fallback (only if ws_size < 512 KiB): white fill + direct red splat.
// ---------------------------------------------------------------------------
__global__ void PR_fill_white(float4* __restrict__ out) {
    unsigned i = blockIdx.x * blockDim.x + threadIdx.x;
    if (i < (unsigned)(HEIGHT * WIDTH)) out[i] = make_float4(1.f, 1.f, 1.f, 1.f);
}

__global__ void PR_splat_direct(const float* __restrict__ centers,
                                const float* __restrict__ vp,
                                int n, float4* __restrict__ out) {
    int i = blockIdx.x * blockDim.x + threadIdx.x;
    if (i >= n) return;
    const float vx = vp[0], vy = vp[1], vw = vp[2], vh = vp[3];
    const float cx = (centers[2 * i]     - vx) / vw * (float)WIDTH;
    const float cy = (float)HEIGHT - (centers[2 * i + 1] - vy) / vh * (float)HEIGHT;
    const int   bx = (int)floorf(cx);
    const int   by = (int)floorf(cy);
    const float r2 = (float)(RADIUS * RADIUS);
    const float4 red = make_float4(1.f, 0.f, 0.f, 1.f);
    for (int dy = -RADIUS; dy <= RADIUS; ++dy) {
        int py = by + dy;
        if (py < 0 || py >= HEIGHT) continue;
        float fy  = (float)py - cy;
        float fy2 = fy * fy;
        for (int dx = -RADIUS; dx <= RADIUS; ++dx) {
            int px = bx + dx;
            if (px < 0 || px >= WIDTH) continue;
            float fx = (float)px - cx;
            if (fx * fx + fy2 <= r2)
                out[(size_t)py * WIDTH + px] = red;  // all writers write red
        }
    }
}

// ---------------------------------------------------------------------------
extern "C" void kernel_launch(void* const* d_in, const int* in_sizes, int n_in,
                              void* d_out, int out_size, void* d_ws, size_t ws_size,
                              hipStream_t stream) {
    (void)n_in; (void)out_size;
    const float* centers = (const float*)d_in[0];   // [N,2] f32
    const float* vp      = (const float*)d_in[1];   // [4]   f32
    const int    n       = in_sizes[0] / 2;
    float4*      out     = (float4*)d_out;          // [H,W,4] f32

    if (d_ws != nullptr && ws_size >= (size_t)MASK_BYTES) {
        unsigned* mask = (unsigned*)d_ws;
        PR_clear_mask<<<(MASK_WORDS / 4 + 255) / 256, 256, 0, stream>>>((uint4*)d_ws);
        PR_splat<<<(n + 255) / 256, 256, 0, stream>>>(centers, vp, n, mask);
        PR_colorize<<<HEIGHT, 256, 0, stream>>>(mask, out);
    } else {
        PR_fill_white<<<(HEIGHT * WIDTH + 255) / 256, 256, 0, stream>>>(out);
        PR_splat_direct<<<(n + 255) / 256, 256, 0, stream>>>(centers, vp, n, out);
    }
}